// GCN_87299505259013
// MI455X (gfx1250) — compile-verified
//
#include <hip/hip_runtime.h>

typedef __attribute__((ext_vector_type(2))) float v2f;
typedef __attribute__((ext_vector_type(8))) float v8f;

// ---------------------------------------------------------------------------
// Degree / normalization
// ---------------------------------------------------------------------------
__global__ void k_deg_init(float* __restrict__ deg, int n) {
  int i = blockIdx.x * blockDim.x + threadIdx.x;
  if (i < n) deg[i] = 1.0f;  // self-loop contribution
}

__global__ void k_deg_count(const int* __restrict__ dst, float* __restrict__ deg, int E) {
  int e = blockIdx.x * blockDim.x + threadIdx.x;
  if (e < E) atomicAdd(&deg[dst[e]], 1.0f);
}

__global__ void k_rsqrt(float* __restrict__ deg, int n) {
  int i = blockIdx.x * blockDim.x + threadIdx.x;
  if (i < n) deg[i] = rsqrtf(deg[i]);  // deg >= 1 always (self loops)
}

// ---------------------------------------------------------------------------
// GEMM1: h1[N,32] = x[N,16] @ W1[16,32]  via V_WMMA_F32_16X16X4_F32
// One wave per 16-row tile; K=16 -> 4 k-steps; 2 column tiles of 16.
// ---------------------------------------------------------------------------
__global__ void k_gemm1_wmma(const float* __restrict__ x, const float* __restrict__ W,
                             float* __restrict__ h, int n, int nTiles) {
  int wave = (int)((blockIdx.x * (unsigned)blockDim.x + threadIdx.x) >> 5);
  if (wave >= nTiles) return;  // wave-uniform; EXEC stays all-ones inside
  int lane = threadIdx.x & 31;
  int half = lane >> 4;   // 0: K offset 0, 1: K offset 2 (A/B 32-bit layout)
  int lan  = lane & 15;   // M for A, N for B/C/D
  int rowBase = wave << 4;
  int row = rowBase + lan;
  int rclamp = row < n ? row : n - 1;  // clamp: loads stay converged
  const float* xrow = x + (size_t)rclamp * 16;

  v8f c0 = {};
  v8f c1 = {};
#pragma unroll
  for (int ks = 0; ks < 4; ++ks) {
    int k0 = ks * 4 + half * 2;
    v2f a  = { xrow[k0], xrow[k0 + 1] };
    v2f b0 = { W[k0 * 32 + lan],      W[(k0 + 1) * 32 + lan] };
    v2f b1 = { W[k0 * 32 + 16 + lan], W[(k0 + 1) * 32 + 16 + lan] };
    c0 = __builtin_amdgcn_wmma_f32_16x16x4_f32(false, a, false, b0, (short)0, c0, false, false);
    c1 = __builtin_amdgcn_wmma_f32_16x16x4_f32(false, a, false, b1, (short)0, c1, false, false);
  }
  if (rowBase + 16 <= n) {           // full tile: uniform branch, unguarded stores
#pragma unroll
    for (int r = 0; r < 8; ++r) {
      int M = rowBase + r + half * 8;
      h[(size_t)M * 32 + lan]      = c0[r];
      h[(size_t)M * 32 + 16 + lan] = c1[r];
    }
  } else {                           // remainder tile only
#pragma unroll
    for (int r = 0; r < 8; ++r) {
      int M = rowBase + r + half * 8;
      if (M < n) {
        h[(size_t)M * 32 + lan]      = c0[r];
        h[(size_t)M * 32 + 16 + lan] = c1[r];
      }
    }
  }
}

// ---------------------------------------------------------------------------
// GEMM2: h2[N,2] = relu(agg1[N,32]) @ W2[32,2], ReLU fused into A-load.
// K=32 -> 8 k-steps; single C tile, only columns 0..1 valid.
// ---------------------------------------------------------------------------
__global__ void k_gemm2_wmma(const float* __restrict__ agg, const float* __restrict__ W,
                             float* __restrict__ h2, int n, int nTiles) {
  int wave = (int)((blockIdx.x * (unsigned)blockDim.x + threadIdx.x) >> 5);
  if (wave >= nTiles) return;
  int lane = threadIdx.x & 31;
  int half = lane >> 4;
  int lan  = lane & 15;
  int rowBase = wave << 4;
  int row = rowBase + lan;
  int rclamp = row < n ? row : n - 1;
  const float* arow = agg + (size_t)rclamp * 32;

  v8f c = {};
#pragma unroll
  for (int ks = 0; ks < 8; ++ks) {
    int k0 = ks * 4 + half * 2;
    v2f a = { fmaxf(arow[k0], 0.0f), fmaxf(arow[k0 + 1], 0.0f) };
    float w0 = (lan < 2) ? W[k0 * 2 + lan]       : 0.0f;
    float w1 = (lan < 2) ? W[(k0 + 1) * 2 + lan] : 0.0f;
    v2f b = { w0, w1 };
    c = __builtin_amdgcn_wmma_f32_16x16x4_f32(false, a, false, b, (short)0, c, false, false);
  }
  if (lan < 2) {
    if (rowBase + 16 <= n) {
#pragma unroll
      for (int r = 0; r < 8; ++r) {
        int M = rowBase + r + half * 8;
        h2[(size_t)M * 2 + lan] = c[r];
      }
    } else {
#pragma unroll
      for (int r = 0; r < 8; ++r) {
        int M = rowBase + r + half * 8;
        if (M < n) h2[(size_t)M * 2 + lan] = c[r];
      }
    }
  }
}

// ---------------------------------------------------------------------------
// agg init (C=32): agg[n][c] = b1[c] + h[n][c]*dis[n]^2, float4 vectorized.
// 8 threads per node.
// ---------------------------------------------------------------------------
__global__ void k_init32(const float* __restrict__ h, const float* __restrict__ dis,
                         const float* __restrict__ bias, float* __restrict__ agg, int n) {
  int t = blockIdx.x * blockDim.x + threadIdx.x;   // n*8 threads
  if (t >= n * 8) return;
  int node = t >> 3;
  int c4 = (t & 7) << 2;
  float s = dis[node];
  float ss = s * s;
  float4 hv = *(const float4*)(h + (size_t)t * 4);
  float4 bv = *(const float4*)(bias + c4);
  float4 o = { bv.x + hv.x * ss, bv.y + hv.y * ss, bv.z + hv.z * ss, bv.w + hv.w * ss };
  *(float4*)(agg + (size_t)t * 4) = o;
}

// agg init (C=2): one thread per node, float2.
__global__ void k_init2(const float* __restrict__ h, const float* __restrict__ dis,
                        const float* __restrict__ bias, float* __restrict__ agg, int n) {
  int node = blockIdx.x * blockDim.x + threadIdx.x;
  if (node >= n) return;
  float s = dis[node];
  float ss = s * s;
  float2 hv = *(const float2*)(h + (size_t)node * 2);
  float2 o = { bias[0] + hv.x * ss, bias[1] + hv.y * ss };
  *(float2*)(agg + (size_t)node * 2) = o;
}

// ---------------------------------------------------------------------------
// Edge scatter (C=32): 8 threads/edge, float4 gather + 4 f32 atomics each.
// agg[dst][c] += h[src][c] * dis[src]*dis[dst]
// ---------------------------------------------------------------------------
__global__ void k_edges32(const int* __restrict__ src, const int* __restrict__ dst,
                          const float* __restrict__ h, const float* __restrict__ dis,
                          float* __restrict__ agg, int E) {
  long long t = (long long)blockIdx.x * blockDim.x + threadIdx.x;  // E*8 threads
  if (t >= (long long)E * 8) return;
  int e = (int)(t >> 3);
  int c4 = ((int)t & 7) << 2;
  int s = src[e];
  int d = dst[e];
  float norm = dis[s] * dis[d];
  float4 hv = *(const float4*)(h + (size_t)s * 32 + c4);
  float* base = agg + (size_t)d * 32 + c4;
  atomicAdd(base + 0, hv.x * norm);
  atomicAdd(base + 1, hv.y * norm);
  atomicAdd(base + 2, hv.z * norm);
  atomicAdd(base + 3, hv.w * norm);
}

// Edge scatter (C=2): 1 thread/edge, float2 gather + 2 f32 atomics.
__global__ void k_edges2(const int* __restrict__ src, const int* __restrict__ dst,
                         const float* __restrict__ h, const float* __restrict__ dis,
                         float* __restrict__ agg, int E) {
  int e = blockIdx.x * blockDim.x + threadIdx.x;
  if (e >= E) return;
  int s = src[e];
  int d = dst[e];
  float norm = dis[s] * dis[d];
  float2 hv = *(const float2*)(h + (size_t)s * 2);
  float* base = agg + (size_t)d * 2;
  atomicAdd(base + 0, hv.x * norm);
  atomicAdd(base + 1, hv.y * norm);
}

// ---------------------------------------------------------------------------
// Host launcher
// Inputs: x[N,16] f32, edge_index[2,E] int, W1[16,32], b1[32], W2[32,2], b2[2]
// Workspace layout (floats): dis[N] | h1[32N] | agg1[32N] | h2[2N]  (~27 MB)
// ---------------------------------------------------------------------------
extern "C" void kernel_launch(void* const* d_in, const int* in_sizes, int n_in,
                              void* d_out, int out_size, void* d_ws, size_t ws_size,
                              hipStream_t stream) {
  (void)n_in; (void)out_size; (void)ws_size;
  const float* x  = (const float*)d_in[0];
  const int*   ei = (const int*)d_in[1];
  const float* W1 = (const float*)d_in[2];
  const float* b1 = (const float*)d_in[3];
  const float* W2 = (const float*)d_in[4];
  const float* b2 = (const float*)d_in[5];
  float* out = (float*)d_out;

  const int N = in_sizes[0] / 16;
  const int E = in_sizes[1] / 2;
  const int* src = ei;        // edge_index[0]
  const int* dst = ei + E;    // edge_index[1]

  float* ws   = (float*)d_ws;
  float* dis  = ws;                         // N
  float* h1   = ws + (size_t)N;             // 32N
  float* agg1 = ws + (size_t)N * 33;        // 32N
  float* h2   = ws + (size_t)N * 65;        // 2N

  const int B = 256;
  // normalization coefficients (shared by both layers)
  k_deg_init<<<(N + B - 1) / B, B, 0, stream>>>(dis, N);
  k_deg_count<<<(E + B - 1) / B, B, 0, stream>>>(dst, dis, E);
  k_rsqrt<<<(N + B - 1) / B, B, 0, stream>>>(dis, N);

  const int nTiles = (N + 15) / 16;
  const int gemmBlocks = (nTiles + 7) / 8;  // 8 waves (wave32) per 256-thread block

  // Layer 1
  k_gemm1_wmma<<<gemmBlocks, B, 0, stream>>>(x, W1, h1, N, nTiles);
  k_init32<<<(N * 8 + B - 1) / B, B, 0, stream>>>(h1, dis, b1, agg1, N);
  {
    long long tot = (long long)E * 8;
    k_edges32<<<(unsigned)((tot + B - 1) / B), B, 0, stream>>>(src, dst, h1, dis, agg1, E);
  }

  // Layer 2 (ReLU fused into GEMM2 A-load)
  k_gemm2_wmma<<<gemmBlocks, B, 0, stream>>>(agg1, W2, h2, N, nTiles);
  k_init2<<<(N + B - 1) / B, B, 0, stream>>>(h2, dis, b2, out, N);
  k_edges2<<<(E + B - 1) / B, B, 0, stream>>>(src, dst, h2, dis, out, E);
}